// GCN_12206297055426
// MI455X (gfx1250) — compile-verified
//
#include <hip/hip_runtime.h>
#include <hip/hip_bf16.h>
#include <math.h>

// Problem constants (match reference)
#define NNODES 100000
#define NEDGES 1600000
#define NLABEL 200000
#define DH_    128
#define DOUT_  64

typedef __attribute__((ext_vector_type(2))) float v2f;
typedef __attribute__((ext_vector_type(8))) float v8f;

// ---------------------------------------------------------------------------
// Degree / normalization
// ---------------------------------------------------------------------------
__global__ void k_init_deg(float* __restrict__ deg, int n) {
  int i = blockIdx.x * blockDim.x + threadIdx.x;
  if (i < n) deg[i] = 1.0f;                       // self-loop contributes 1
}

__global__ void k_accum_deg(const int* __restrict__ dst, float* __restrict__ deg, int e) {
  int i = blockIdx.x * blockDim.x + threadIdx.x;
  if (i < e) atomicAdd(&deg[dst[i]], 1.0f);
}

__global__ void k_dis(float* __restrict__ deg_dis, int n) {
  int i = blockIdx.x * blockDim.x + threadIdx.x;
  if (i < n) deg_dis[i] = rsqrtf(deg_dis[i]);     // deg >= 1 always (self-loop)
}

__global__ void k_norm(const int* __restrict__ src, const int* __restrict__ dst,
                       const float* __restrict__ dis, float* __restrict__ nrm, int e) {
  int i = blockIdx.x * blockDim.x + threadIdx.x;
  if (i < e) nrm[i] = dis[src[i]] * dis[dst[i]];
}

// ---------------------------------------------------------------------------
// GEMM: H[N x D] = X[N x 128] @ W[128 x D], fp32 via V_WMMA_F32_16X16X4_F32.
// Each wave owns ONE column tile (ct) for its whole lifetime: the 32 B
// fragments (64 VGPRs) are loaded from global exactly once, then reused
// across all row tiles. Inner loop = 32 x (global_load_b64 A + WMMA); no LDS,
// no barriers, loads freely pipelined by the compiler.
// Optionally applies ReLU to the input on load (fuses the activation of the
// previous layer's aggregation output).
// N, D multiples of 16; all control flow wave-uniform -> EXEC all-ones.
// ---------------------------------------------------------------------------
template <bool RELU_IN>
__global__ void k_gemm_wmma(const float* __restrict__ X, const float* __restrict__ W,
                            float* __restrict__ H, int nRowTiles, int ctShift) {
  const int lane   = threadIdx.x & 31;
  const int wave   = threadIdx.x >> 5;
  const int wpb    = blockDim.x >> 5;
  const int gwave  = blockIdx.x * wpb + wave;
  const int nwaves = gridDim.x * wpb;

  const int nColTiles = 1 << ctShift;             // 8 (D=128) or 4 (D=64)
  const int D   = nColTiles << 4;
  const int ct  = gwave & (nColTiles - 1);
  const int r0  = gwave >> ctShift;
  const int rstride = nwaves >> ctShift;          // nwaves is multiple of nColTiles

  const int l16  = lane & 15;
  const int hi   = lane >> 4;                     // 0: lanes 0-15, 1: lanes 16-31
  const int col0 = ct << 4;

  // B 4x16 fragment layout: VGPR0 lanes0-15=K0 / lanes16-31=K1; VGPR1 K2/K3.
  // Loaded once per wave, reused for every row tile.
  v2f Bf[32];
#pragma unroll
  for (int kk = 0; kk < 32; ++kk) {
    Bf[kk].x = W[(4 * kk + hi) * D + col0 + l16];
    Bf[kk].y = W[(4 * kk + 2 + hi) * D + col0 + l16];
  }

  for (int rt = r0; rt < nRowTiles; rt += rstride) {   // wave-uniform loop
    // A 16x4 layout: lanes 0-15 hold K=k,k+1; lanes 16-31 hold K=k+2,k+3
    const float* xrow = X + (size_t)(rt * 16 + l16) * 128 + 2 * hi;
    v8f acc = {0.f, 0.f, 0.f, 0.f, 0.f, 0.f, 0.f, 0.f};
#pragma unroll
    for (int kk = 0; kk < 32; ++kk) {
      v2f a = *(const v2f*)(xrow + 4 * kk);       // 8B-aligned global_load_b64
      if (RELU_IN) { a.x = fmaxf(a.x, 0.f); a.y = fmaxf(a.y, 0.f); }
      acc = __builtin_amdgcn_wmma_f32_16x16x4_f32(
          /*neg_a=*/false, a, /*neg_b=*/false, Bf[kk],
          /*c_mod=*/(short)0, acc, /*reuse_a=*/false, /*reuse_b=*/false);
    }
    // C/D layout: VGPR r -> M=r (lanes 0-15), M=r+8 (lanes 16-31), N=l16
    float* hrow = H + (size_t)(rt * 16 + 8 * hi) * D + col0 + l16;
#pragma unroll
    for (int r = 0; r < 8; ++r) hrow[(size_t)r * D] = acc[r];
  }
}

// ---------------------------------------------------------------------------
// OUT[i][d] = bias[d]  (bias is added AFTER aggregation in GCNConv)
// ---------------------------------------------------------------------------
__global__ void k_init_bias(float* __restrict__ out, const float* __restrict__ b,
                            int total, int dMask /* D-1, D is pow2 */) {
  int i = blockIdx.x * blockDim.x + threadIdx.x;
  if (i < total) out[i] = b[i & dMask];
}

// ---------------------------------------------------------------------------
// Edge scatter: OUT[dst] += w * H[src]; one wave per edge. Each lane owns one
// float4 (D=128) or float2 (D=64): one vector load + scalar f32 atomics.
// Edges [0,E) use norm[]; [E, E+N) are the self-loops with w = dis[i]^2.
// Feature table (51.2MB) is L2-resident -> gathers + atomics hit the 192MB L2.
// ---------------------------------------------------------------------------
template <int DFEAT>
__global__ void k_scatter(const float* __restrict__ H, float* __restrict__ OUT,
                          const int* __restrict__ src, const int* __restrict__ dst,
                          const float* __restrict__ nrm, const float* __restrict__ dis,
                          int e, int n) {
  constexpr int VEC = DFEAT / 32;                 // 4 or 2
  const int lane = threadIdx.x & 31;
  const int wave = (blockIdx.x * blockDim.x + threadIdx.x) >> 5;
  const int nwaves = (gridDim.x * blockDim.x) >> 5;
  const int total = e + n;
  for (int t = wave; t < total; t += nwaves) {    // wave-uniform
    int s, d; float w;
    if (t < e) { s = src[t]; d = dst[t]; w = nrm[t]; }
    else       { s = d = t - e; float di = dis[s]; w = di * di; }
    const float* hs = H + (size_t)s * DFEAT + lane * VEC;
    float* od = OUT + (size_t)d * DFEAT + lane * VEC;
    if constexpr (VEC == 4) {
      float4 v = *(const float4*)hs;              // global_load_b128
      atomicAdd(od + 0, w * v.x);
      atomicAdd(od + 1, w * v.y);
      atomicAdd(od + 2, w * v.z);
      atomicAdd(od + 3, w * v.w);
    } else {
      float2 v = *(const float2*)hs;              // global_load_b64
      atomicAdd(od + 0, w * v.x);
      atomicAdd(od + 1, w * v.y);
    }
  }
}

// ---------------------------------------------------------------------------
// Decode: hidden = z[a]*z[b]; logits = sum; hidden_n = hidden/max(||hidden||,1e-12)
// One wave per label edge; each lane owns 2 of the 64 dims; wave32 xor-reduce.
// ---------------------------------------------------------------------------
__global__ void k_decode(const float* __restrict__ Z, const int* __restrict__ eli,
                         float* __restrict__ hidden, float* __restrict__ logits, int el) {
  const int lane = threadIdx.x & 31;
  const int wave = (blockIdx.x * blockDim.x + threadIdx.x) >> 5;
  const int nwaves = (gridDim.x * blockDim.x) >> 5;
  for (int l = wave; l < el; l += nwaves) {
    const int a = eli[l];
    const int b = eli[el + l];
    const float* za = Z + (size_t)a * DOUT_;
    const float* zb = Z + (size_t)b * DOUT_;
    float h0 = za[lane]      * zb[lane];
    float h1 = za[32 + lane] * zb[32 + lane];
    float s = h0 + h1;
    float q = h0 * h0 + h1 * h1;
#pragma unroll
    for (int off = 16; off > 0; off >>= 1) {
      s += __shfl_xor(s, off, 32);
      q += __shfl_xor(q, off, 32);
    }
    float nm = fmaxf(sqrtf(q), 1e-12f);
    hidden[(size_t)l * DOUT_ + lane]      = h0 / nm;
    hidden[(size_t)l * DOUT_ + 32 + lane] = h1 / nm;
    if (lane == 0) logits[l] = s;
  }
}

// ---------------------------------------------------------------------------
extern "C" void kernel_launch(void* const* d_in, const int* in_sizes, int n_in,
                              void* d_out, int out_size, void* d_ws, size_t ws_size,
                              hipStream_t stream) {
  (void)in_sizes; (void)n_in; (void)out_size; (void)ws_size;

  const float* x  = (const float*)d_in[0];
  const float* W1 = (const float*)d_in[1];
  const float* b1 = (const float*)d_in[2];
  const float* W2 = (const float*)d_in[3];
  const float* b2 = (const float*)d_in[4];
  const float* W3 = (const float*)d_in[5];
  const float* b3 = (const float*)d_in[6];
  const int*   ei  = (const int*)d_in[7];   // [2, E] flat: row0 = src, row1 = dst
  const int*   eli = (const int*)d_in[8];   // [2, EL]
  const int* srcE = ei;
  const int* dstE = ei + NEDGES;

  // Workspace layout (floats): dis[N] | norm[E] | B0[N*128] | B1[N*128]  (~110MB)
  float* ws   = (float*)d_ws;
  float* dis  = ws;
  float* nrm  = dis + NNODES;
  float* B0   = nrm + NEDGES;
  float* B1   = B0 + (size_t)NNODES * DH_;

  const int T = 256;
  const int GEMM_BLOCKS = 1024;   // 8192 waves; multiple of nColTiles (8 and 4)
  const int SCAT_BLOCKS = 4096;   // grid-stride over 1.7M edges
  const int nNDH = NNODES * DH_;
  const int nNDO = NNODES * DOUT_;

  // --- symmetric normalization ---
  k_init_deg <<<(NNODES + T - 1) / T, T, 0, stream>>>(dis, NNODES);
  k_accum_deg<<<(NEDGES + T - 1) / T, T, 0, stream>>>(dstE, dis, NEDGES);
  k_dis      <<<(NNODES + T - 1) / T, T, 0, stream>>>(dis, NNODES);
  k_norm     <<<(NEDGES + T - 1) / T, T, 0, stream>>>(srcE, dstE, dis, nrm, NEDGES);

  // --- layer 1: H = x@W1 -> B0 ; B1 = b1 + scatter(H) ---
  k_gemm_wmma<false><<<GEMM_BLOCKS, T, 0, stream>>>(x, W1, B0, NNODES / 16, 3);
  k_init_bias<<<(nNDH + T - 1) / T, T, 0, stream>>>(B1, b1, nNDH, DH_ - 1);
  k_scatter<DH_><<<SCAT_BLOCKS, T, 0, stream>>>(B0, B1, srcE, dstE, nrm, dis, NEDGES, NNODES);

  // --- layer 2 (ReLU of B1 fused into the GEMM A-load) ---
  k_gemm_wmma<true><<<GEMM_BLOCKS, T, 0, stream>>>(B1, W2, B0, NNODES / 16, 3);
  k_init_bias<<<(nNDH + T - 1) / T, T, 0, stream>>>(B1, b2, nNDH, DH_ - 1);
  k_scatter<DH_><<<SCAT_BLOCKS, T, 0, stream>>>(B0, B1, srcE, dstE, nrm, dis, NEDGES, NNODES);

  // --- layer 3 (ReLU fused; no ReLU on layer-3 output) ---
  k_gemm_wmma<true><<<GEMM_BLOCKS, T, 0, stream>>>(B1, W3, B0, NNODES / 16, 2);
  k_init_bias<<<(nNDO + T - 1) / T, T, 0, stream>>>(B1, b3, nNDO, DOUT_ - 1);
  k_scatter<DOUT_><<<SCAT_BLOCKS, T, 0, stream>>>(B0, B1, srcE, dstE, nrm, dis, NEDGES, NNODES);

  // --- decode: d_out = [hidden_n (EL*64) | logits (EL)] ---
  float* hid = (float*)d_out;
  float* lgt = hid + (size_t)NLABEL * DOUT_;
  k_decode<<<4096, T, 0, stream>>>(B1, eli, hid, lgt, NLABEL);
}